// TagLoss_87136296501782
// MI455X (gfx1250) — compile-verified
//
#include <hip/hip_runtime.h>
#include <math.h>

#define NUM_G 64
#define DIM 8
#define SLOTS 10      // 8 dim-sums + count + sum(|p|^2)
#define GSTRIDE 11    // padded LDS stride: gcd(11,64)=1 -> group bases hit all banks
#define BLOCK 256
#define WAVES 8
#define CHUNKS 160
#define BIMG 8
#define EPSF 1e-6f

typedef float v2f __attribute__((ext_vector_type(2)));
typedef float v8f __attribute__((ext_vector_type(8)));

// ---------------- zero workspace ----------------
__global__ void tag_zero_kernel(float* __restrict__ p, int n) {
    int i = blockIdx.x * blockDim.x + threadIdx.x;
    if (i < n) p[i] = 0.0f;
}

// ---------------- streaming one-pass accumulation ----------------
// acc layout: [img][g*SLOTS + s], s: 0..7 = sum pred[d], 8 = count, 9 = sum |p|^2
__global__ void __launch_bounds__(BLOCK)
tag_accum_kernel(const float* __restrict__ pred, const int* __restrict__ gt,
                 float* __restrict__ acc, int N) {
    __shared__ float lacc[WAVES * NUM_G * GSTRIDE];
    const int tid  = threadIdx.x;
    const int wave = tid >> 5;

    for (int s = tid; s < WAVES * NUM_G * GSTRIDE; s += BLOCK) lacc[s] = 0.0f;
    __syncthreads();

    const int img = blockIdx.y;
    const float4* __restrict__ p4 = (const float4*)(pred + (size_t)img * N * DIM);
    const int* __restrict__ gti = gt + (size_t)img * N;
    float* wacc = lacc + wave * (NUM_G * GSTRIDE);

    const int stride = gridDim.x * BLOCK;
    for (int i = blockIdx.x * BLOCK + tid; i < N; i += stride) {
        float4 a = p4[2 * i];
        float4 b = p4[2 * i + 1];
        int g = gti[i];
        __builtin_prefetch(&p4[2 * (i + stride)], 0, 0);   // global_prefetch_b8
        float sq = a.x*a.x + a.y*a.y + a.z*a.z + a.w*a.w
                 + b.x*b.x + b.y*b.y + b.z*b.z + b.w*b.w;
        float* slot = wacc + g * GSTRIDE;
        atomicAdd(slot + 0, a.x);   // ds_add_f32 (no-return)
        atomicAdd(slot + 1, a.y);
        atomicAdd(slot + 2, a.z);
        atomicAdd(slot + 3, a.w);
        atomicAdd(slot + 4, b.x);
        atomicAdd(slot + 5, b.y);
        atomicAdd(slot + 6, b.z);
        atomicAdd(slot + 7, b.w);
        atomicAdd(slot + 8, 1.0f);
        atomicAdd(slot + 9, sq);
    }
    __syncthreads();

    // combine the 8 per-wave copies, one native FP32 global atomic per slot
    float* gacc = acc + img * (NUM_G * SLOTS);
    for (int s = tid; s < NUM_G * SLOTS; s += BLOCK) {
        int g = s / SLOTS, off = s % SLOTS;
        float v = 0.0f;
        #pragma unroll
        for (int w = 0; w < WAVES; ++w)
            v += lacc[w * (NUM_G * GSTRIDE) + g * GSTRIDE + off];
        unsafeAtomicAdd(gacc + s, v);                      // global_atomic_add_f32
    }
}

// ---------------- finalize: means, pull, WMMA Gram -> push ----------------
// one block, 8 waves; wave w handles image w.
__global__ void __launch_bounds__(256)
tag_finalize_kernel(const float* __restrict__ acc, float* __restrict__ out) {
    __shared__ float smeans[BIMG][NUM_G * DIM];  // 16 KB
    __shared__ float ssq[BIMG][NUM_G];
    __shared__ float spres[BIMG][NUM_G];
    __shared__ float sloss[BIMG];

    const int tid  = threadIdx.x;
    const int wave = tid >> 5;
    const int lane = tid & 31;
    const int img  = wave;
    const float* a = acc + img * (NUM_G * SLOTS);

    // means, presence, pull via  sum|p|^2 - c*|m|^2
    float pullsum = 0.0f, numf = 0.0f;
    for (int g = lane; g < NUM_G; g += 32) {
        float cnt   = a[g * SLOTS + 8];
        float sumsq = a[g * SLOTS + 9];
        float sc    = fmaxf(cnt, 1.0f);
        float msq   = 0.0f;
        #pragma unroll
        for (int d = 0; d < DIM; ++d) {
            float m = a[g * SLOTS + d] / sc;
            smeans[img][g * DIM + d] = m;
            msq += m * m;
        }
        ssq[img][g] = msq;
        float pres = (cnt > 0.0f) ? 1.0f : 0.0f;
        spres[img][g] = pres;
        pullsum += pres * (sumsq - cnt * msq) / (sc * (float)DIM);
        numf    += pres;
    }
    #pragma unroll
    for (int off = 16; off > 0; off >>= 1) {
        pullsum += __shfl_xor(pullsum, off, 32);
        numf    += __shfl_xor(numf, off, 32);
    }

    // Gram = M * M^T  (64x64, K=8) with v_wmma_f32_16x16x4_f32.
    // Operand layout (ISA 7.12.2, 32-bit): lanes 0-15 carry K=kb+{0,1} in the
    // two VGPRs, lanes 16-31 carry K=kb+{2,3}; A row / B col = lane%16.
    const int row   = lane & 15;
    const int khalf = lane >> 4;
    float pushsum = 0.0f;
    for (int ti = 0; ti < 4; ++ti) {
        for (int tj = 0; tj < 4; ++tj) {
            v8f c = {};
            #pragma unroll
            for (int kb = 0; kb < DIM; kb += 4) {
                int k0 = kb + 2 * khalf;
                v2f A, Bv;
                A.x  = smeans[img][(ti * 16 + row) * DIM + k0];
                A.y  = smeans[img][(ti * 16 + row) * DIM + k0 + 1];
                Bv.x = smeans[img][(tj * 16 + row) * DIM + k0];
                Bv.y = smeans[img][(tj * 16 + row) * DIM + k0 + 1];
                c = __builtin_amdgcn_wmma_f32_16x16x4_f32(
                        false, A, false, Bv, (short)0, c, false, false);
            }
            // C element (M = v + 8*khalf, N = row)
            #pragma unroll
            for (int v = 0; v < 8; ++v) {
                int gi = ti * 16 + v + 8 * khalf;
                int gj = tj * 16 + row;
                float d2 = ssq[img][gi] + ssq[img][gj] - 2.0f * c[v];
                pushsum += __expf(-d2) * spres[img][gi] * spres[img][gj];
            }
        }
    }
    #pragma unroll
    for (int off = 16; off > 0; off >>= 1) pushsum += __shfl_xor(pushsum, off, 32);

    if (lane == 0) {
        float num  = numf;
        float pull = pullsum / (num + EPSF);
        float push = (pushsum - num) / ((num - 1.0f) * num + EPSF) * 0.5f;
        sloss[img] = push + pull;   // PUSH_WEIGHT = PULL_WEIGHT = 1
    }
    __syncthreads();
    if (tid == 0) {
        float t = 0.0f;
        #pragma unroll
        for (int b = 0; b < BIMG; ++b) t += sloss[b];
        out[0] = t * (1.0f / (float)BIMG);   // LOSS_WEIGHT = 1
    }
}

extern "C" void kernel_launch(void* const* d_in, const int* in_sizes, int n_in,
                              void* d_out, int out_size, void* d_ws, size_t ws_size,
                              hipStream_t stream) {
    const float* pred = (const float*)d_in[0];
    const int*   gt   = (const int*)d_in[1];
    // d_in[2] (anchor_inds) unused on the group_anchor=False path.
    const int N = in_sizes[1] / BIMG;   // elements per image

    float* acc = (float*)d_ws;          // BIMG * 64 * 10 floats = 20 KB
    const int accN = BIMG * NUM_G * SLOTS;

    tag_zero_kernel<<<(accN + 255) / 256, 256, 0, stream>>>(acc, accN);

    dim3 grid(CHUNKS, BIMG);
    tag_accum_kernel<<<grid, BLOCK, 0, stream>>>(pred, gt, acc, N);

    tag_finalize_kernel<<<1, 256, 0, stream>>>(acc, (float*)d_out);
}